// LSTM_UNIT_25898652794991
// MI455X (gfx1250) — compile-verified
//
#include <hip/hip_runtime.h>
#include <math.h>

typedef float v2f __attribute__((ext_vector_type(2)));
typedef float v4f __attribute__((ext_vector_type(4)));
typedef float v8f __attribute__((ext_vector_type(8)));

#define HDIM 4096
#define VDIM 32000

// Accumulate 16 row-dot-products of W[rowbase..rowbase+15][0:4096] with the
// vector staged in LDS, using V_WMMA_F32_16X16X4_F32.
//
// All 16 rows of the A fragment carry the SAME vector chunk (no masking
// needed: every row of D then computes the same dots; we consume row 0 only,
// which lives in VGPR c[0], lanes 0..15, N = lane).
//
// K-permuted mapping so one b128 load feeds two WMMAs:
//   WMMA#0 K-slots <-> cols {k, k+1, k+4, k+5}
//   WMMA#1 K-slots <-> cols {k+2, k+3, k+6, k+7}
// f32 operand layout (ISA 7.12.2): lane L holds K-pair 2*(L>>4) of row (L&15),
// so lane-half 0 loads cols k..k+3 and half 1 loads cols k+4..k+7.
__device__ __forceinline__ void mv16_accum(const float* __restrict__ W,
                                           const float* __restrict__ vlds,
                                           int rowbase, v8f& c0, v8f& c1) {
  const int lane = threadIdx.x & 31;
  const int r    = lane & 15;
  const int hoff = (lane >> 4) << 2;                 // 0 or 4
  const float* wp = W + (size_t)(rowbase + r) * HDIM + hoff;
  const float* vp = vlds + hoff;
#pragma unroll 4
  for (int k = 0; k < HDIM; k += 8) {
    v4f b = *(const v4f*)(wp + k);   // W[row][k+hoff .. k+hoff+3]
    v4f a = *(const v4f*)(vp + k);   // vec[k+hoff .. k+hoff+3] (broadcast)
    v2f a0; a0.x = a.x; a0.y = a.y;
    v2f b0; b0.x = b.x; b0.y = b.y;
    c0 = __builtin_amdgcn_wmma_f32_16x16x4_f32(false, a0, false, b0,
                                               (short)0, c0, false, false);
    v2f a1; a1.x = a.z; a1.y = a.w;
    v2f b1; b1.x = b.z; b1.y = b.w;
    c1 = __builtin_amdgcn_wmma_f32_16x16x4_f32(false, a1, false, b1,
                                               (short)0, c1, false, false);
  }
}

// gates_raw[g*4096 + row] = dot(Wx_g[row], x) + dot(Wh_g[row], h)
// 128 blocks x 256 threads = 1024 waves; wave -> (gate = w>>8, rowtile = w&255)
__global__ void __launch_bounds__(256) gates_matvec(
    const float* __restrict__ wxf, const float* __restrict__ whf,
    const float* __restrict__ wxi, const float* __restrict__ whi,
    const float* __restrict__ wxg, const float* __restrict__ whg,
    const float* __restrict__ wxo, const float* __restrict__ who,
    const float* __restrict__ x, const float* __restrict__ h,
    float* __restrict__ gates_raw) {
  __shared__ __align__(16) float sx[HDIM];
  __shared__ __align__(16) float sh[HDIM];
  for (int i = threadIdx.x; i < HDIM; i += 256) {
    sx[i] = x[i];
    sh[i] = h[i];
  }
  __syncthreads();

  const int wave    = blockIdx.x * 8 + (threadIdx.x >> 5);
  const int gate    = wave >> 8;                  // 0:f 1:i 2:g 3:o
  const int rowbase = (wave & 255) << 4;          // 0..4080
  const float* Wx = (gate == 0) ? wxf : (gate == 1) ? wxi : (gate == 2) ? wxg : wxo;
  const float* Wh = (gate == 0) ? whf : (gate == 1) ? whi : (gate == 2) ? whg : who;

  v8f c0 = {};
  v8f c1 = {};
  mv16_accum(Wx, sx, rowbase, c0, c1);
  mv16_accum(Wh, sh, rowbase, c0, c1);

  const int lane = threadIdx.x & 31;
  if (lane < 16)
    gates_raw[gate * HDIM + rowbase + lane] = c0[0] + c1[0];
}

// f/i/g/o activations, c_t, h_t.  4096 threads.
__global__ void __launch_bounds__(256) lstm_pointwise(
    const float* __restrict__ gates_raw,
    const float* __restrict__ bf, const float* __restrict__ bi,
    const float* __restrict__ bg, const float* __restrict__ bo,
    const float* __restrict__ prev_c,
    float* __restrict__ out,        // [0:4096) h_t, [4096:8192) c_t
    float* __restrict__ h_ws) {
  const int r = blockIdx.x * blockDim.x + threadIdx.x;
  const float zf = gates_raw[r]            + bf[r];
  const float zi = gates_raw[HDIM + r]     + bi[r];
  const float zg = gates_raw[2 * HDIM + r] + bg[r];
  const float zo = gates_raw[3 * HDIM + r] + bo[r];
  const float f = 1.0f / (1.0f + expf(-zf));
  const float i = 1.0f / (1.0f + expf(-zi));
  const float g = tanhf(zg);
  const float o = 1.0f / (1.0f + expf(-zo));
  const float c = f * prev_c[r] + i * g;
  const float ht = o * tanhf(c);
  out[r]        = ht;
  out[HDIM + r] = c;
  h_ws[r]       = ht;
}

// logits[row] = dot(w_hy[row], h_t) + b_y[row].  250 blocks x 8 waves = 32000 rows.
__global__ void __launch_bounds__(256) vocab_matvec(
    const float* __restrict__ why, const float* __restrict__ hvec,
    const float* __restrict__ by, float* __restrict__ logits) {
  __shared__ __align__(16) float shv[HDIM];
  for (int i = threadIdx.x; i < HDIM; i += 256) shv[i] = hvec[i];
  __syncthreads();

  const int wave    = blockIdx.x * 8 + (threadIdx.x >> 5);
  const int rowbase = wave << 4;

  v8f c0 = {};
  v8f c1 = {};
  mv16_accum(why, shv, rowbase, c0, c1);

  const int lane = threadIdx.x & 31;
  if (lane < 16)
    logits[rowbase + lane] = c0[0] + c1[0] + by[rowbase + lane];
}

// Deterministic single-block softmax over 32000 logits.
__global__ void __launch_bounds__(1024) softmax_k(const float* __restrict__ logits,
                                                  float* __restrict__ y) {
  __shared__ float red[1024];
  const int tid = threadIdx.x;

  float m = -3.402823466e38f;
  for (int i = tid; i < VDIM; i += 1024) m = fmaxf(m, logits[i]);
  red[tid] = m;
  __syncthreads();
  for (int s = 512; s > 0; s >>= 1) {
    if (tid < s) red[tid] = fmaxf(red[tid], red[tid + s]);
    __syncthreads();
  }
  const float mx = red[0];
  __syncthreads();

  float sum = 0.0f;
  for (int i = tid; i < VDIM; i += 1024) sum += expf(logits[i] - mx);
  red[tid] = sum;
  __syncthreads();
  for (int s = 512; s > 0; s >>= 1) {
    if (tid < s) red[tid] += red[tid + s];
    __syncthreads();
  }
  const float inv = 1.0f / red[0];

  for (int i = tid; i < VDIM; i += 1024) y[i] = expf(logits[i] - mx) * inv;
}

extern "C" void kernel_launch(void* const* d_in, const int* in_sizes, int n_in,
                              void* d_out, int out_size, void* d_ws, size_t ws_size,
                              hipStream_t stream) {
  const float* prev_h = (const float*)d_in[0];
  const float* prev_c = (const float*)d_in[1];
  const float* x_t    = (const float*)d_in[2];
  const float* w_xf   = (const float*)d_in[3];
  const float* w_hf   = (const float*)d_in[4];
  const float* b_f    = (const float*)d_in[5];
  const float* w_xi   = (const float*)d_in[6];
  const float* w_hi   = (const float*)d_in[7];
  const float* b_i    = (const float*)d_in[8];
  const float* w_xg   = (const float*)d_in[9];
  const float* w_hg   = (const float*)d_in[10];
  const float* b_g    = (const float*)d_in[11];
  const float* w_xo   = (const float*)d_in[12];
  const float* w_ho   = (const float*)d_in[13];
  const float* b_o    = (const float*)d_in[14];
  const float* w_hy   = (const float*)d_in[15];
  const float* b_y    = (const float*)d_in[16];

  float* out = (float*)d_out;            // [h_t(4096) | c_t(4096) | y_t(32000)]
  float* ws  = (float*)d_ws;
  float* gates_raw = ws;                 // 16384 floats
  float* h_ws      = ws + 4 * HDIM;      // 4096 floats
  float* logits    = ws + 5 * HDIM;      // 32000 floats

  gates_matvec<<<128, 256, 0, stream>>>(w_xf, w_hf, w_xi, w_hi,
                                        w_xg, w_hg, w_xo, w_ho,
                                        x_t, prev_h, gates_raw);
  lstm_pointwise<<<HDIM / 256, 256, 0, stream>>>(gates_raw, b_f, b_i, b_g, b_o,
                                                 prev_c, out, h_ws);
  vocab_matvec<<<VDIM / 128, 256, 0, stream>>>(w_hy, h_ws, b_y, logits);
  softmax_k<<<1, 1024, 0, stream>>>(logits, out + 2 * HDIM);
}